// CenterLoss_19361712571337
// MI455X (gfx1250) — compile-verified
//
#include <hip/hip_runtime.h>
#include <hip/hip_bf16.h>

typedef __attribute__((ext_vector_type(2))) float v2f;
typedef __attribute__((ext_vector_type(8))) float v8f;

#define FEAT_DIM   2048
#define BATCH      1024
#define ROWS_PER_BLK 16
#define KT         512            // K-tile width (floats) staged in LDS
#define LDS_STRIDE 516            // 512 + 4 pad: stride % 64 == 4 -> conflict-free b64 reads
#define CLAMP_MIN  1e-12f
#define CLAMP_MAX  1e12f

// CDNA5 async copy: global -> LDS, no VGPR round trip, tracked by ASYNCcnt.
__device__ __forceinline__ void async_ld_b128(unsigned lds_off, const float* g) {
    asm volatile("global_load_async_to_lds_b128 %0, %1, off"
                 :: "v"(lds_off), "v"((unsigned long long)(uintptr_t)g)
                 : "memory");
}

__device__ __forceinline__ void wait_async0() {
#if __has_builtin(__builtin_amdgcn_s_wait_asynccnt)
    __builtin_amdgcn_s_wait_asynccnt(0);
#else
    asm volatile("s_wait_asynccnt 0x0" ::: "memory");
#endif
}

// Kernel 1: per-(row, K-half) partial sums of squared diff via WMMA diagonal trick.
// grid = (BATCH/16, 2), block = 256 (8 waves of 32).
__global__ __launch_bounds__(256)
void centerloss_partial_kernel(const float* __restrict__ x,
                               const int* __restrict__ labels,
                               const float* __restrict__ centers,
                               float* __restrict__ ws) {
    __shared__ __align__(16) float xs[ROWS_PER_BLK * LDS_STRIDE]; // ~33 KB
    __shared__ __align__(16) float cs[ROWS_PER_BLK * LDS_STRIDE]; // ~33 KB
    __shared__ float part[8 * ROWS_PER_BLK];

    const int t    = threadIdx.x;
    const int lane = t & 31;
    const int w    = t >> 5;
    const int r0   = blockIdx.x * ROWS_PER_BLK;
    const int kh   = blockIdx.y;              // K-half: 0 or 1

    // Phase-A assignment: one row per 16 threads -> label & base pointers are
    // loop-invariant, loaded exactly once. 16 consecutive lanes cover a 256 B
    // contiguous segment per async b128 batch.
    const int row = t >> 4;
    const int c0  = t & 15;
    const float* xrow = x + (size_t)(r0 + row) * FEAT_DIM;
    const float* crow = centers + (size_t)labels[r0 + row] * FEAT_DIM;
    const unsigned lds_elem = (unsigned)(row * LDS_STRIDE + c0 * 4) * 4u; // bytes
    const unsigned lx = (unsigned)(uintptr_t)(void*)xs + lds_elem;
    const unsigned lc = (unsigned)(uintptr_t)(void*)cs + lds_elem;

    // Phase-B operand addressing (A == B by WMMA f32 layout symmetry):
    // lanes 0-15 supply rows m with K = k0,k0+1 ; lanes 16-31 supply K = k0+2,k0+3.
    const int m    = lane & 15;
    const int koff = (lane >> 4) * 2;
    const float* xb = &xs[m * LDS_STRIDE + w * 64 + koff];
    const float* cb = &cs[m * LDS_STRIDE + w * 64 + koff];

    v8f acc0 = {};
    v8f acc1 = {};

    for (int tile = 0; tile < 2; ++tile) {
        const int kbase = kh * (FEAT_DIM / 2) + tile * KT;

        // ---- Phase A: 16 async b128 copies per thread, all in flight ----
        const float* gx = xrow + kbase + c0 * 4;
        const float* gc = crow + kbase + c0 * 4;
        #pragma unroll
        for (int i = 0; i < 8; ++i) {
            async_ld_b128(lx + (unsigned)i * 256u, gx + i * 64);
            async_ld_b128(lc + (unsigned)i * 256u, gc + i * 64);
        }
        wait_async0();
        __syncthreads();

        // ---- Phase B: wave w owns K-slice [w*64, w*64+64) of the tile ----
        #pragma unroll
        for (int i = 0; i < 16; i += 2) {
            float2 xa0 = *(const float2*)(xb + (i + 0) * 4);
            float2 ca0 = *(const float2*)(cb + (i + 0) * 4);
            float2 xa1 = *(const float2*)(xb + (i + 1) * 4);
            float2 ca1 = *(const float2*)(cb + (i + 1) * 4);
            v2f av0 = {xa0.x - ca0.x, xa0.y - ca0.y};
            v2f av1 = {xa1.x - ca1.x, xa1.y - ca1.y};
            // D = A x A^T + C : diagonal accumulates per-row sum of squares
            acc0 = __builtin_amdgcn_wmma_f32_16x16x4_f32(
                false, av0, false, av0, (short)0, acc0, false, false);
            acc1 = __builtin_amdgcn_wmma_f32_16x16x4_f32(
                false, av1, false, av1, (short)0, acc1, false, false);
        }
        __syncthreads();   // before next tile overwrites the staging buffers
    }

    // Diagonal of 16x16 f32 C/D: element (m,m):
    //   m in [0,8): VGPR m at lane m ; m in [8,16): VGPR m-8 at lane m+16.
    float dval = 0.0f;
    int   prow = -1;
    #pragma unroll
    for (int j = 0; j < 8; ++j) {
        if (lane == j)      { dval = acc0[j] + acc1[j]; prow = j; }
        if (lane == j + 24) { dval = acc0[j] + acc1[j]; prow = j + 8; }
    }
    if (prow >= 0) part[w * ROWS_PER_BLK + prow] = dval;
    __syncthreads();

    if (t < ROWS_PER_BLK) {
        float s = 0.0f;
        #pragma unroll
        for (int ww = 0; ww < 8; ++ww) s += part[ww * ROWS_PER_BLK + t];
        ws[kh * BATCH + r0 + t] = s;   // unscaled, unclamped partial
    }
}

// Kernel 2: combine K-halves, scale, clamp, mean -> d_out[0]
__global__ __launch_bounds__(256)
void centerloss_reduce_kernel(const float* __restrict__ ws,
                              float* __restrict__ out) {
    __shared__ float s[256];
    const int t = threadIdx.x;
    float acc = 0.0f;
    for (int r = t; r < BATCH; r += 256) {
        float d = (ws[r] + ws[BATCH + r]) * (1.0f / (float)FEAT_DIM);
        d = fminf(fmaxf(d, CLAMP_MIN), CLAMP_MAX);
        acc += d;
    }
    s[t] = acc;
    __syncthreads();
    for (int off = 128; off > 0; off >>= 1) {
        if (t < off) s[t] += s[t + off];
        __syncthreads();
    }
    if (t == 0) out[0] = s[0] * (1.0f / (float)BATCH);
}

extern "C" void kernel_launch(void* const* d_in, const int* in_sizes, int n_in,
                              void* d_out, int out_size, void* d_ws, size_t ws_size,
                              hipStream_t stream) {
    const float* x       = (const float*)d_in[0];
    const int*   labels  = (const int*)d_in[1];
    const float* centers = (const float*)d_in[2];
    float* out = (float*)d_out;
    float* ws  = (float*)d_ws;   // needs 2 * 1024 floats = 8 KB

    dim3 grid(BATCH / ROWS_PER_BLK, 2);
    centerloss_partial_kernel<<<grid, 256, 0, stream>>>(x, labels, centers, ws);
    centerloss_reduce_kernel<<<1, 256, 0, stream>>>(ws, out);
}